// DigitConvolutionalModel_72911364817621
// MI455X (gfx1250) — compile-verified
//
#include <hip/hip_runtime.h>
#include <stdint.h>
#include <stddef.h>

// Problem constants (match the JAX reference)
#define BATCH 32768
#define IMG   28
#define OUTW  26          // IMG - 3 + 1
#define FLAT  676         // OUTW * OUTW
#define NCLS  10

// Tiling
#define WAVES 4                       // waves per block
#define TPB   (WAVES * 32)            // 128 threads
#define SPW   16                      // samples per wave (WMMA M)
#define SPB   (WAVES * SPW)           // 64 samples per block
#define CHUNK_ROWS 2
#define CHUNK_K    (CHUNK_ROWS * OUTW)    // 52 h-values per chunk
#define NCHUNKS    (OUTW / CHUNK_ROWS)    // 13

// LDS carve (floats)
#define SWT_FLOATS ((FLAT / 2) * 32)      // 10816: fc_w transposed + K-pair interleaved
#define SH_FLOATS  (SPW * CHUNK_K)        // 832 per wave: h tile for current chunk
#define XR_FLOATS  (SPW * 4 * IMG)        // 1792 per wave: ring buffer of 4 image rows
#define SMEM_BYTES ((SWT_FLOATS + WAVES * SH_FLOATS + WAVES * XR_FLOATS) * 4)  // 85248

typedef __attribute__((ext_vector_type(2))) float v2f;
typedef __attribute__((ext_vector_type(8))) float v8f;

// Async bulk copy global -> LDS (16B), tracked with ASYNCcnt. GV mode:
//   global_load_async_to_lds_b128 vdst(lds byte offset), vaddr(64b), off
__device__ __forceinline__ void async_copy_b128(const float* gsrc, const float* ldst) {
    uint32_t lds_off = (uint32_t)(uintptr_t)ldst;   // flat LDS addr[31:0] = LDS byte offset
    asm volatile("global_load_async_to_lds_b128 %0, %1, off"
                 :: "v"(lds_off), "v"(gsrc) : "memory");
}
__device__ __forceinline__ void wait_async_zero() {
    asm volatile("s_wait_asynccnt 0x0" ::: "memory");
}
__device__ __forceinline__ void wait_ds_zero() {
    asm volatile("s_wait_dscnt 0x0" ::: "memory");
}

// Fused conv3x3(valid) + ReLU + FC(676->10), single pass over x.
// FC via V_WMMA_F32_16X16X4_F32: per wave M=16 samples, N=16 (10 classes
// zero-padded), K over 676 in steps of 4. Image rows stream through a
// per-wave 4-row LDS ring via GLOBAL_LOAD_ASYNC_TO_LDS_B128 (1x HBM traffic).
__global__ __launch_bounds__(TPB)
void digit_conv_fc_kernel(const float* __restrict__ x,
                          const float* __restrict__ conv_w,
                          const float* __restrict__ fc_w,
                          const float* __restrict__ fc_b,
                          float* __restrict__ out)
{
    extern __shared__ float smem[];
    float* sWt = smem;                                         // 43264 B
    float* hT  = smem + SWT_FLOATS;                            // per-wave below
    float* xr  = smem + SWT_FLOATS + WAVES * SH_FLOATS;        // per-wave below

    const int tid  = threadIdx.x;
    const int wave = tid >> 5;
    const int lane = tid & 31;
    const int half = lane >> 4;   // 0: lanes 0-15, 1: lanes 16-31
    const int n    = lane & 15;   // B/C/D column (class, padded to 16)
    const int m    = lane & 15;   // A row (sample within wave tile)

    hT += wave * SH_FLOATS;
    float* xw = xr + wave * XR_FLOATS;   // [16 samples][4 ring rows][28 floats]

    const int samp_base = blockIdx.x * SPB + wave * SPW;

    // 3x3 conv weights in registers (uniform)
    const float w00 = conv_w[0], w01 = conv_w[1], w02 = conv_w[2];
    const float w10 = conv_w[3], w11 = conv_w[4], w12 = conv_w[5];
    const float w20 = conv_w[6], w21 = conv_w[7], w22 = conv_w[8];

    // Prologue: async-stage image rows 0..3 for this wave's 16 samples.
    // 16 samples x 4 rows x 28 floats = 448 b128 transfers, 14 per lane.
#pragma unroll
    for (int i = 0; i < 14; ++i) {
        const int idx = lane + i * 32;           // < 448
        const int s = idx / 28;                  // sample
        const int t = idx % 28;                  // vec4 within rows 0..3
        async_copy_b128(x + (size_t)(samp_base + s) * (IMG * IMG) + t * 4,
                        xw + s * (4 * IMG) + t * 4);
    }

    // Stage FC weights transposed + K-pair interleaved:
    //   sWt[(k>>1)*32 + n*2 + (k&1)] = (n < 10) ? fc_w[n*676 + k] : 0
    for (int idx = tid; idx < FLAT * 16; idx += TPB) {
        const int k  = idx >> 4;
        const int nn = idx & 15;
        const float v = (nn < NCLS) ? fc_w[nn * FLAT + k] : 0.0f;
        sWt[((k >> 1) << 5) + (nn << 1) + (k & 1)] = v;
    }
    __syncthreads();   // sWt visible to all waves (the only cross-wave sharing)

    // Accumulator initialized with bias (depends on column n only).
    const float bias = (n < NCLS) ? fc_b[n] : 0.0f;
    v8f acc;
#pragma unroll
    for (int i = 0; i < 8; ++i) acc[i] = bias;

    for (int chunk = 0; chunk < NCHUNKS; ++chunk) {
        const int row0 = chunk * CHUNK_ROWS;     // first output row of chunk

        // Staged rows (ring slots (row0..row0+3)&3) must have landed.
        wait_async_zero();

        // ---- conv + ReLU: 2 output rows x 16 samples -> hT[16][52] ----
#pragma unroll 1
        for (int s = 0; s < SPW; ++s) {
            const float* bs = xw + s * (4 * IMG);
#pragma unroll
            for (int it = 0; it < 2; ++it) {     // 52 positions / 32 lanes
                const int p = lane + it * 32;
                if (p < CHUNK_K) {
                    const int oy = row0 + (p >= OUTW ? 1 : 0);
                    const int ox = (p >= OUTW) ? (p - OUTW) : p;
                    const float* r0 = bs + ((oy + 0) & 3) * IMG + ox;
                    const float* r1 = bs + ((oy + 1) & 3) * IMG + ox;
                    const float* r2 = bs + ((oy + 2) & 3) * IMG + ox;
                    float a = r0[0] * w00 + r0[1] * w01 + r0[2] * w02
                            + r1[0] * w10 + r1[1] * w11 + r1[2] * w12
                            + r2[0] * w20 + r2[1] * w21 + r2[2] * w22;
                    hT[s * CHUNK_K + p] = a > 0.0f ? a : 0.0f;
                }
            }
        }

        // All conv LDS stores (and ring reads) retired before reuse below.
        wait_ds_zero();

        // ---- async-stage rows row0+4, row0+5 for next chunk (overlaps WMMA).
        // They overwrite ring slots of rows row0, row0+1 (dead after this conv).
        if (chunk + 1 < NCHUNKS) {
            const int rnew = row0 + 4;                    // even; slots rnew&3, +1
#pragma unroll
            for (int i = 0; i < 7; ++i) {
                const int idx = lane + i * 32;            // < 224
                const int s = idx / 14;                   // sample
                const int t = idx % 14;                   // vec4 within the 2 rows
                async_copy_b128(
                    x + (size_t)(samp_base + s) * (IMG * IMG) + rnew * IMG + t * 4,
                    xw + s * (4 * IMG) + (rnew & 3) * IMG + t * 4);
            }
        }

        // ---- WMMA: 13 steps of K=4 over this chunk's 52 h values ----
#pragma unroll
        for (int step = 0; step < CHUNK_K / 4; ++step) {
            const int kloc  = step * 4 + half * 2;        // K within chunk (even)
            const int kglob = chunk * CHUNK_K + kloc;     // global K (even)
            // A 16x4 f32: VGPR0 = h[m][kbase+2*half], VGPR1 = +1
            const v2f a = *(const v2f*)&hT[m * CHUNK_K + kloc];
            // B 4x16 f32: VGPR0 = Wt[kbase+2*half][n], VGPR1 = +1
            const v2f b = *(const v2f*)&sWt[((kglob >> 1) << 5) + (n << 1)];
            acc = __builtin_amdgcn_wmma_f32_16x16x4_f32(
                /*neg_a=*/false, a, /*neg_b=*/false, b,
                /*c_mod=*/(short)0, acc, /*reuse_a=*/false, /*reuse_b=*/false);
        }
    }

    // ---- writeout: D layout — VGPR i: lanes 0-15 -> M=i, lanes 16-31 -> M=8+i
    if (n < NCLS) {
#pragma unroll
        for (int i = 0; i < 8; ++i) {
            const int mrow = i + half * 8;
            out[(size_t)(samp_base + mrow) * NCLS + n] = acc[i];
        }
    }
}

extern "C" void kernel_launch(void* const* d_in, const int* in_sizes, int n_in,
                              void* d_out, int out_size, void* d_ws, size_t ws_size,
                              hipStream_t stream) {
    (void)in_sizes; (void)n_in; (void)d_ws; (void)ws_size; (void)out_size;
    const float* x      = (const float*)d_in[0];
    const float* conv_w = (const float*)d_in[1];
    const float* fc_w   = (const float*)d_in[2];
    const float* fc_b   = (const float*)d_in[3];
    float* out = (float*)d_out;

    dim3 grid(BATCH / SPB);   // 512 blocks
    dim3 block(TPB);          // 128 threads = 4 waves
    digit_conv_fc_kernel<<<grid, block, SMEM_BYTES, stream>>>(x, conv_w, fc_w, fc_b, out);
}